// SpectralConv3d_fast_1211180777829
// MI455X (gfx1250) — compile-verified
//
#include <hip/hip_runtime.h>
#include <hip/hip_bf16.h>

// SpectralConv3d (FNO) on gfx1250 via factorized partial DFTs, all stages as
// V_WMMA_F32_16X16X4_F32 GEMMs. Memory-bound (~0.6 GB HBM @ 23.3 TB/s floor),
// so f32 WMMA matches reference precision at no cost.
//
// Workspace layout (floats), total 23,068,672 floats = 92.3 MB:
//   S1 [0,       16777216)  fwd-T  : [b][ci][x][m=(c,kt)16][part2][y64]
//   S2 [16777216,20971520)  fwd-Y  : [b][ci][ky16][m16][part2][x64]
//   S3 [20971520,22020096)  fwd-X  : [kx][ky][kt][c][b4][part2][ci32]
//   S4 [22020096,23068672)  mixed  : [b][co][ky16][m16][part2][kx16]
//   S5 [0,        4194304)  inv-X  : [b][co][x64][m16][part2][ky16]   (reuse S1)
//   S6 [4194304, 20971520)  inv-Y  : [b][co][x64][c2][y64][kt8 re|8 im] (reuse)

typedef __attribute__((ext_vector_type(2))) float v2f;
typedef __attribute__((ext_vector_type(8))) float v8f;

#define TWO_PI 6.28318530717958647692f

__device__ __forceinline__ v8f wmma4(v2f a, v2f b, v8f c) {
  // D = A(16x4,f32) * B(4x16,f32) + C(16x16,f32)
  return __builtin_amdgcn_wmma_f32_16x16x4_f32(false, a, false, b, (short)0, c,
                                               false, false);
}

// LDS twiddle layout: rows paired so B operand (rows k,k+1 at col n) is one
// contiguous 8-byte ds_load.
__device__ __forceinline__ int lpk(int k, int n, int ncols2) {
  return (k >> 1) * ncols2 + n * 2 + (k & 1);  // ncols2 = 2*ncols
}

// ---------------- Stage 1: forward partial DFT over T (real -> 8 cplx modes)
// One wave per (b,ci,x,c,yblock). A[m=y16][k=t], B[t][n] = n<8: cos/64,
// n>=8: -sin/64. D[y][kt re|im] stored with y innermost for stage 2.
__global__ void k1_dft_t(const float* __restrict__ x, float* __restrict__ S1) {
  __shared__ __align__(16) float Wt[64 * 16];
  const int tid = threadIdx.x;
  for (int i = tid; i < 64 * 16; i += 256) {
    int k = i >> 4, n = i & 15;
    float ang = TWO_PI * (float)(((n & 7) * k) & 63) * (1.0f / 64.0f);
    Wt[lpk(k, n, 32)] =
        (n < 8) ? cosf(ang) * (1.0f / 64.0f) : -sinf(ang) * (1.0f / 64.0f);
  }
  __syncthreads();
  const int w = blockIdx.x * 8 + (tid >> 5);
  const int lo = tid & 15, hi = (tid >> 4) & 1;
  const int yb = w & 3, c = (w >> 2) & 1, xx = (w >> 3) & 63;
  const int ci = (w >> 9) & 31, b = w >> 14;
  const float* xa = x + ((b * 32 + ci) * 64 + xx) * 8192;  // [y][t][c]
  const int y = yb * 16 + lo;
  v8f acc = {};
  for (int s = 0; s < 16; ++s) {
    int k0 = s * 4 + hi * 2;
    v2f a;
    a.x = xa[(y * 64 + k0) * 2 + c];
    a.y = xa[(y * 64 + k0 + 1) * 2 + c];
    v2f bb = *(const v2f*)&Wt[(k0 >> 1) * 32 + lo * 2];
    acc = wmma4(a, bb, acc);
  }
  // col n=lo -> kt=lo&7, part=lo>>3; rows y = yb*16 + 8*hi + r (contiguous)
  float* o = S1 + (((b * 32 + ci) * 64 + xx) * 16 + (c * 8 + (lo & 7))) * 128 +
             (lo >> 3) * 64 + yb * 16 + hi * 8;
  *(float4*)&o[0] = make_float4(acc[0], acc[1], acc[2], acc[3]);
  *(float4*)&o[4] = make_float4(acc[4], acc[5], acc[6], acc[7]);
}

// Shared LDS builder for the forward complex 64->16-mode DFT matrices,
// K split [re(64) | im(64)] :  B1 = [Wre; -Wim],  B2 = [Wim; Wre].
__device__ __forceinline__ void build_fwd_tables(float* B1, float* B2) {
  for (int i = threadIdx.x; i < 128 * 16; i += 256) {
    int k = i >> 4, n = i & 15;
    int sp = k & 63;
    bool im = k >= 64;
    int f = (n < 8) ? n : (n + 48);  // modes 0..7, 56..63
    float ang = TWO_PI * (float)((f * sp) & 63) * (1.0f / 64.0f);
    float wre = cosf(ang) * (1.0f / 64.0f);
    float wim = -sinf(ang) * (1.0f / 64.0f);
    int li = lpk(k, n, 32);
    B1[li] = im ? -wim : wre;
    B2[li] = im ? wre : wim;
  }
}

// ---------------- Stage 2: forward DFT over Y (complex, 64 -> 16 modes)
// One wave per (b,ci,x). A[m=(c,kt)][k=re|im over y] contiguous from S1.
__global__ void k2_dft_y(const float* __restrict__ S1, float* __restrict__ S2) {
  __shared__ __align__(16) float B1[128 * 16], B2[128 * 16];
  build_fwd_tables(B1, B2);
  __syncthreads();
  const int tid = threadIdx.x;
  const int w = blockIdx.x * 8 + (tid >> 5);
  const int lo = tid & 15, hi = (tid >> 4) & 1;
  const int xx = w & 63, ci = (w >> 6) & 31, b = w >> 11;
  const float* A = S1 + ((b * 32 + ci) * 64 + xx) * 2048;
  v8f aR = {}, aI = {};
  for (int s = 0; s < 32; ++s) {
    int k0 = s * 4 + hi * 2;
    v2f a = *(const v2f*)&A[lo * 128 + k0];
    v2f b1 = *(const v2f*)&B1[(k0 >> 1) * 32 + lo * 2];
    v2f b2 = *(const v2f*)&B2[(k0 >> 1) * 32 + lo * 2];
    aR = wmma4(a, b1, aR);
    aI = wmma4(a, b2, aI);
  }
  int base = ((b * 32 + ci) * 16 + lo) * 2048 + hi * 1024 + xx;  // ky = lo
  for (int r = 0; r < 8; ++r) {
    S2[base + r * 128] = aR[r];
    S2[base + r * 128 + 64] = aI[r];
  }
}

// ---------------- Stage 3: forward DFT over X (complex, 64 -> 16 modes)
__global__ void k3_dft_x(const float* __restrict__ S2, float* __restrict__ S3) {
  __shared__ __align__(16) float B1[128 * 16], B2[128 * 16];
  build_fwd_tables(B1, B2);
  __syncthreads();
  const int tid = threadIdx.x;
  const int w = blockIdx.x * 8 + (tid >> 5);
  const int lo = tid & 15, hi = (tid >> 4) & 1;
  const int ky = w & 15, ci = (w >> 4) & 31, b = w >> 9;
  const float* A = S2 + ((b * 32 + ci) * 16 + ky) * 2048;
  v8f aR = {}, aI = {};
  for (int s = 0; s < 32; ++s) {
    int k0 = s * 4 + hi * 2;
    v2f a = *(const v2f*)&A[lo * 128 + k0];
    v2f b1 = *(const v2f*)&B1[(k0 >> 1) * 32 + lo * 2];
    v2f b2 = *(const v2f*)&B2[(k0 >> 1) * 32 + lo * 2];
    aR = wmma4(a, b1, aR);
    aI = wmma4(a, b2, aI);
  }
  for (int r = 0; r < 8; ++r) {  // kx = lo (col), m = r+8*hi -> (c,kt)
    int m = hi * 8 + r;
    int idx = ((((lo * 16 + ky) * 8 + (m & 7)) * 2 + (m >> 3)) * 4 + b) * 64 + ci;
    S3[idx] = aR[r];
    S3[idx + 32] = aI[r];
  }
}

// ---------------- Stage 4: per-mode complex Cin->Cout mix (einsum over ci)
// One wave per (mode=(kx,ky,kt,c), co-tile). M rows 0..3 = batch b.
__global__ void k4_modemix(const float* __restrict__ S3,
                           const float* __restrict__ w1,
                           const float* __restrict__ w2,
                           const float* __restrict__ w3,
                           const float* __restrict__ w4,
                           float* __restrict__ S4) {
  const int tid = threadIdx.x;
  const int w = blockIdx.x * 8 + (tid >> 5);
  const int lo = tid & 15, hi = (tid >> 4) & 1;
  const int cotile = w & 1, c = (w >> 1) & 1, kt = (w >> 2) & 7;
  const int ky = (w >> 5) & 15, kx = (w >> 9) & 15;
  const float* wsel;
  int m1, m2;
  if (kx < 8) {
    m1 = kx;
    if (ky < 8) { wsel = w1; m2 = ky; } else { wsel = w3; m2 = ky - 8; }
  } else {
    m1 = kx - 8;
    if (ky < 8) { wsel = w2; m2 = ky; } else { wsel = w4; m2 = ky - 8; }
  }
  const int co = cotile * 16 + lo;
  const int wbase = 2 * (co * 1024 + m1 * 128 + m2 * 16 + kt * 2 + c);
  const int abase = (((kx * 16 + ky) * 8 + kt) * 2 + c) * 256;
  const int brow = lo & 3;  // wrap M rows 4..15 onto valid batch rows
  v8f aR = {}, aI = {};
  for (int s = 0; s < 16; ++s) {
    int k0 = s * 4 + hi * 2;  // even; rows (k0,k0+1) stay in same re/im half
    int part = k0 >> 5, cidx = k0 & 31;
    v2f a = *(const v2f*)&S3[abase + brow * 64 + part * 32 + cidx];
    v2f cw0 = *(const v2f*)&wsel[wbase + cidx * 65536];
    v2f cw1 = *(const v2f*)&wsel[wbase + (cidx + 1) * 65536];
    bool re_half = (k0 < 32);  // uniform across lanes per step
    v2f b1, b2;
    b1.x = re_half ? cw0.x : -cw0.y;
    b1.y = re_half ? cw1.x : -cw1.y;
    b2.x = re_half ? cw0.y : cw0.x;
    b2.y = re_half ? cw1.y : cw1.x;
    aR = wmma4(a, b1, aR);
    aI = wmma4(a, b2, aI);
  }
  if (hi == 0) {
    for (int r = 0; r < 4; ++r) {  // b = r
      int idx = (((r * 32 + co) * 16 + ky) * 16 + (c * 8 + kt)) * 32 + kx;
      S4[idx] = aR[r];
      S4[idx + 16] = aI[r];
    }
  }
}

// Inverse complex 16-mode -> 64-point tables (no 1/N: norm='forward' inverse):
// B1 = [cos; -sin], B2 = [sin; cos], 64 spatial columns per row.
__device__ __forceinline__ void build_inv_tables(float* B1, float* B2) {
  for (int i = threadIdx.x; i < 32 * 64; i += 256) {
    int k = i >> 6, sp = i & 63;
    int kk = k & 15;
    bool im = k >= 16;
    int f = (kk < 8) ? kk : kk + 48;
    float ang = TWO_PI * (float)((f * sp) & 63) * (1.0f / 64.0f);
    float cr = cosf(ang), si = sinf(ang);
    int li = (k >> 1) * 128 + sp * 2 + (k & 1);
    B1[li] = im ? -si : cr;
    B2[li] = im ? cr : si;
  }
}

// ---------------- Stage 5: inverse DFT over X (16 modes -> 64 points)
__global__ void k5_idft_x(const float* __restrict__ S4, float* __restrict__ S5) {
  __shared__ __align__(16) float B1[32 * 64], B2[32 * 64];
  build_inv_tables(B1, B2);
  __syncthreads();
  const int tid = threadIdx.x;
  const int w = blockIdx.x * 8 + (tid >> 5);
  const int lo = tid & 15, hi = (tid >> 4) & 1;
  const int xb = w & 3, ky = (w >> 2) & 15, co = (w >> 6) & 31, b = w >> 11;
  const float* A = S4 + ((b * 32 + co) * 16 + ky) * 512;
  const int col = xb * 16 + lo;
  v8f aR = {}, aI = {};
  for (int s = 0; s < 8; ++s) {
    int k0 = s * 4 + hi * 2;
    v2f a = *(const v2f*)&A[lo * 32 + k0];
    v2f b1 = *(const v2f*)&B1[(k0 >> 1) * 128 + col * 2];
    v2f b2 = *(const v2f*)&B2[(k0 >> 1) * 128 + col * 2];
    aR = wmma4(a, b1, aR);
    aI = wmma4(a, b2, aI);
  }
  int base = (((b * 32 + co) * 64 + col) * 16 + hi * 8) * 32 + ky;
  for (int r = 0; r < 8; ++r) {
    S5[base + r * 32] = aR[r];
    S5[base + r * 32 + 16] = aI[r];
  }
}

// ---------------- Stage 6: inverse DFT over Y (16 modes -> 64 points)
__global__ void k6_idft_y(const float* __restrict__ S5, float* __restrict__ S6) {
  __shared__ __align__(16) float B1[32 * 64], B2[32 * 64];
  build_inv_tables(B1, B2);
  __syncthreads();
  const int tid = threadIdx.x;
  const int w = blockIdx.x * 8 + (tid >> 5);
  const int lo = tid & 15, hi = (tid >> 4) & 1;
  const int yb = w & 3, x = (w >> 2) & 63, co = (w >> 8) & 31, b = w >> 13;
  const float* A = S5 + ((b * 32 + co) * 64 + x) * 512;
  const int col = yb * 16 + lo;
  v8f aR = {}, aI = {};
  for (int s = 0; s < 8; ++s) {
    int k0 = s * 4 + hi * 2;
    v2f a = *(const v2f*)&A[lo * 32 + k0];
    v2f b1 = *(const v2f*)&B1[(k0 >> 1) * 128 + col * 2];
    v2f b2 = *(const v2f*)&B2[(k0 >> 1) * 128 + col * 2];
    aR = wmma4(a, b1, aR);
    aI = wmma4(a, b2, aI);
  }
  for (int r = 0; r < 8; ++r) {  // y = col, m = r+8*hi -> (c,kt)
    int m = hi * 8 + r;
    int idx = ((((b * 32 + co) * 64 + x) * 2 + (m >> 3)) * 64 + col) * 16 + (m & 7);
    S6[idx] = aR[r];
    S6[idx + 8] = aI[r];
  }
}

// ---------------- Stage 7: inverse rDFT over T (8 cplx modes -> 64 reals)
// x[t] = Re(Z0) + sum_{kt=1..7} 2*(Zr cos - Zi sin);  kt=0 imag dropped.
__global__ void k7_idft_t(const float* __restrict__ S6, float* __restrict__ out) {
  __shared__ __align__(16) float Bt[16 * 64];
  for (int i = threadIdx.x; i < 16 * 64; i += 256) {
    int k = i >> 6, t = i & 63;
    float v;
    if (k < 8) {
      float ang = TWO_PI * (float)((k * t) & 63) * (1.0f / 64.0f);
      v = ((k == 0) ? 1.0f : 2.0f) * cosf(ang);
    } else {
      int kt = k - 8;
      float ang = TWO_PI * (float)((kt * t) & 63) * (1.0f / 64.0f);
      v = (kt == 0) ? 0.0f : -2.0f * sinf(ang);
    }
    Bt[(k >> 1) * 128 + t * 2 + (k & 1)] = v;
  }
  __syncthreads();
  const int tid = threadIdx.x;
  const int w = blockIdx.x * 8 + (tid >> 5);
  const int lo = tid & 15, hi = (tid >> 4) & 1;
  const int tb = w & 3, yb = (w >> 2) & 3, c = (w >> 4) & 1;
  const int x = (w >> 5) & 63, co = (w >> 11) & 31, b = w >> 16;
  const float* A = S6 + (((b * 32 + co) * 64 + x) * 2 + c) * 1024;
  const int y = yb * 16 + lo;
  const int t = tb * 16 + lo;
  v8f acc = {};
  for (int s = 0; s < 4; ++s) {
    int k0 = s * 4 + hi * 2;
    v2f a = *(const v2f*)&A[y * 16 + k0];
    v2f bb = *(const v2f*)&Bt[(k0 >> 1) * 128 + t * 2];
    acc = wmma4(a, bb, acc);
  }
  int obase = ((((b * 32 + co) * 64 + x) * 64 + yb * 16 + hi * 8) * 64 + t) * 2 + c;
  for (int r = 0; r < 8; ++r) out[obase + r * 128] = acc[r];
}

extern "C" void kernel_launch(void* const* d_in, const int* in_sizes, int n_in,
                              void* d_out, int out_size, void* d_ws,
                              size_t ws_size, hipStream_t stream) {
  const float* x = (const float*)d_in[0];
  const float* w1 = (const float*)d_in[1];
  const float* w2 = (const float*)d_in[2];
  const float* w3 = (const float*)d_in[3];
  const float* w4 = (const float*)d_in[4];
  float* out = (float*)d_out;
  float* ws = (float*)d_ws;

  float* S1 = ws;              // 16,777,216 f
  float* S2 = ws + 16777216;   //  4,194,304 f
  float* S3 = ws + 20971520;   //  1,048,576 f
  float* S4 = ws + 22020096;   //  1,048,576 f
  float* S5 = ws;              //  4,194,304 f (reuse: S1 dead)
  float* S6 = ws + 4194304;    // 16,777,216 f (reuse: S1 tail + S2 dead)

  k1_dft_t<<<8192, 256, 0, stream>>>(x, S1);
  k2_dft_y<<<1024, 256, 0, stream>>>(S1, S2);
  k3_dft_x<<<256, 256, 0, stream>>>(S2, S3);
  k4_modemix<<<1024, 256, 0, stream>>>(S3, w1, w2, w3, w4, S4);
  k5_idft_x<<<1024, 256, 0, stream>>>(S4, S5);
  k6_idft_y<<<4096, 256, 0, stream>>>(S5, S6);
  k7_idft_t<<<32768, 256, 0, stream>>>(S6, out);
}